// LinearNodes_85143431676073
// MI455X (gfx1250) — compile-verified
//
#include <hip/hip_runtime.h>

#define N_TOKENS  16384
#define NUM_NODES 64
#define IN_DIM    128
#define OUT_DIM   128
#define MAX_TILES 1088      // <= N_TOKENS/16 + NUM_NODES partial tiles
#define AS_STRIDE 132       // 128 + 4 pad: stride%64 banks = 4 -> conflict-free

typedef __attribute__((ext_vector_type(2))) float v2f;
typedef __attribute__((ext_vector_type(8))) float v8f;

// ---------------- setup kernels (MoE dispatch) ----------------

__global__ void ln_zero(int* __restrict__ p) {
    p[threadIdx.x] = 0;   // counts[64] + cursors[64]
}

__global__ void ln_hist(const int* __restrict__ node, int* __restrict__ counts) {
    int t = blockIdx.x * blockDim.x + threadIdx.x;
    if (t < N_TOKENS) atomicAdd(&counts[node[t] & (NUM_NODES - 1)], 1);
}

__global__ void ln_scan_build(const int* __restrict__ counts, int* __restrict__ offsets,
                              int* __restrict__ numTiles, int* __restrict__ tileE,
                              int* __restrict__ tileStart, int* __restrict__ tileValid) {
    if (threadIdx.x == 0 && blockIdx.x == 0) {
        int off = 0, nt = 0;
        for (int e = 0; e < NUM_NODES; ++e) {
            offsets[e] = off;
            int c = counts[e];
            for (int s = 0; s < c; s += 16) {
                tileE[nt]     = e;
                tileStart[nt] = off + s;
                int v = c - s;
                tileValid[nt] = v < 16 ? v : 16;
                ++nt;
            }
            off += c;
        }
        offsets[NUM_NODES] = off;
        *numTiles = nt;
    }
}

__global__ void ln_scatter(const int* __restrict__ node, const int* __restrict__ offsets,
                           int* __restrict__ cursors, int* __restrict__ bucket) {
    int t = blockIdx.x * blockDim.x + threadIdx.x;
    if (t < N_TOKENS) {
        int e = node[t] & (NUM_NODES - 1);
        int p = atomicAdd(&cursors[e], 1);
        bucket[offsets[e] + p] = t;
    }
}

// ---------------- WMMA GEMM: one 16-token tile x 128 outputs per block ----------------

__global__ __launch_bounds__(256) void ln_wmma_gemm(
    const float* __restrict__ x, const float* __restrict__ weight,
    const float* __restrict__ bias, float* __restrict__ out,
    const int* __restrict__ numTiles, const int* __restrict__ tileE,
    const int* __restrict__ tileStart, const int* __restrict__ tileValid,
    const int* __restrict__ bucket) {

    __shared__ float As[16 * AS_STRIDE];
    __shared__ int   tokIds[16];

    int bx = blockIdx.x;
    if (bx >= *numTiles) return;   // uniform early exit for overprovisioned grid

    const int e     = tileE[bx];
    const int start = tileStart[bx];
    const int valid = tileValid[bx];
    const int tid   = threadIdx.x;

    if (tid < 16) tokIds[tid] = (tid < valid) ? bucket[start + tid] : -1;
    __syncthreads();

    // Stage A: gather 16 token rows (16 x 128 f32) into LDS, float4 at a time.
    for (int q = tid; q < 16 * 32; q += 256) {
        int row = q >> 5;
        int c4  = q & 31;
        int tok = tokIds[row];
        float4 v = make_float4(0.f, 0.f, 0.f, 0.f);
        if (tok >= 0) v = reinterpret_cast<const float4*>(x)[tok * 32 + c4];
        *reinterpret_cast<float4*>(&As[row * AS_STRIDE + c4 * 4]) = v;
    }
    __syncthreads();

    const int lane = tid & 31;
    const int wid  = tid >> 5;            // wave id 0..7 -> 16-wide output column tile
    const int col  = lane & 15;           // N in tile; also A row per WMMA A layout
    const int kb   = (lane >> 4) * 2;     // K sub-offset per A/B lane layout
    const int nOut = wid * 16 + col;

    // B[k][n] = weight[e][n][k]; lane holds {B[kb][col], B[kb+1][col]}
    const float* wrow = weight + ((size_t)e * OUT_DIM + nOut) * IN_DIM;
    const float* arow = &As[col * AS_STRIDE];

    // C/D element (m,n): vgpr j holds m = j + (lane>>4)*8, n = lane&15.
    // Bias depends on n only -> broadcast into all 8 accumulator regs.
    const float bv = bias[e * OUT_DIM + nOut];
    v8f c;
#pragma unroll
    for (int j = 0; j < 8; ++j) c[j] = bv;

#pragma unroll
    for (int k = 0; k < IN_DIM; k += 4) {
        v2f a = *reinterpret_cast<const v2f*>(&arow[k + kb]);   // LDS, 8B, conflict-free
        v2f b = *reinterpret_cast<const v2f*>(&wrow[k + kb]);   // global (L2-resident)
        c = __builtin_amdgcn_wmma_f32_16x16x4_f32(
                /*neg_a=*/false, a, /*neg_b=*/false, b,
                /*c_mod=*/(short)0, c, /*reuse_a=*/false, /*reuse_b=*/false);
    }

    const int mBase = (lane >> 4) * 8;
#pragma unroll
    for (int j = 0; j < 8; ++j) {
        int tok = tokIds[mBase + j];
        if (tok >= 0) out[(size_t)tok * OUT_DIM + nOut] = c[j];
    }
}

// ---------------- launch ----------------

extern "C" void kernel_launch(void* const* d_in, const int* in_sizes, int n_in,
                              void* d_out, int out_size, void* d_ws, size_t ws_size,
                              hipStream_t stream) {
    const float* x      = (const float*)d_in[0];
    const int*   node   = (const int*)d_in[1];
    const float* weight = (const float*)d_in[2];
    const float* bias   = (const float*)d_in[3];
    float*       out    = (float*)d_out;

    int* counts    = (int*)d_ws;            // 64
    int* cursors   = counts    + 64;        // 64
    int* offsets   = cursors   + 64;        // 65
    int* numTiles  = offsets   + 65;        // 1
    int* tileE     = numTiles  + 1;         // MAX_TILES
    int* tileStart = tileE     + MAX_TILES; // MAX_TILES
    int* tileValid = tileStart + MAX_TILES; // MAX_TILES
    int* bucket    = tileValid + MAX_TILES; // N_TOKENS

    ln_zero<<<1, 128, 0, stream>>>(counts);                       // counts + cursors
    ln_hist<<<N_TOKENS / 256, 256, 0, stream>>>(node, counts);
    ln_scan_build<<<1, 32, 0, stream>>>(counts, offsets, numTiles, tileE, tileStart, tileValid);
    ln_scatter<<<N_TOKENS / 256, 256, 0, stream>>>(node, offsets, cursors, bucket);
    ln_wmma_gemm<<<MAX_TILES, 256, 0, stream>>>(x, weight, bias, out,
                                                numTiles, tileE, tileStart, tileValid, bucket);
}